// GATLayer_14637248544973
// MI455X (gfx1250) — compile-verified
//
#include <hip/hip_runtime.h>
#include <math.h>

#define KNODES   100000
#define EEDGES   1600000
#define IN_FEAT  256
#define NHEADS   4
#define HDIM     32
#define OUT_F    128           // NHEADS * HDIM
#define NEG_SLOPE 0.2f

typedef __attribute__((ext_vector_type(2))) float v2f;
typedef __attribute__((ext_vector_type(8))) float v8f;

// -----------------------------------------------------------------------------
// Kernel 1: Wh[K,128] = x[K,256] @ W^T   (W is [128,256] row-major)
// Wave32 WMMA f32 16x16x4. Each wave: 16 rows x full 128 cols (8 N-tiles).
// W staged per-block in LDS in 64-wide K chunks, padded stride 68 words
// (bank = (4n + kk + 2h) % 64 -> conflict-free b64 B-fragment loads).
// A/B fragments are software-pipelined one k-step ahead so WMMAs consume
// registers that are already resident instead of draining DScnt each step.
// -----------------------------------------------------------------------------
__global__ __launch_bounds__(256) void gat_gemm_wmma(
    const float* __restrict__ x, const float* __restrict__ W,
    float* __restrict__ Wh)
{
    __shared__ float Wlds[128 * 68];

    const int tid    = threadIdx.x;
    const int wave   = tid >> 5;
    const int lane   = tid & 31;
    const int lane16 = lane & 15;
    const int half   = lane >> 4;          // 0: lanes 0-15, 1: lanes 16-31

    const int m0   = blockIdx.x * 128 + wave * 16;
    int       mRow = m0 + lane16;
    int       mCl  = mRow < KNODES ? mRow : (KNODES - 1);   // clamp, keep EXEC full
    const float* xrow = x + (size_t)mCl * IN_FEAT;

    v8f acc[8];
#pragma unroll
    for (int nt = 0; nt < 8; ++nt) acc[nt] = (v8f){};

    for (int kc = 0; kc < IN_FEAT; kc += 64) {
        __syncthreads();
        // Stage W[:,kc:kc+64] into LDS: 2048 float4 loads, 8 per thread, coalesced.
#pragma unroll
        for (int i = 0; i < 8; ++i) {
            int lin = tid + 256 * i;
            int n   = lin >> 4;            // 0..127 output column
            int q   = lin & 15;            // float4 index within chunk row
            float4 wv = *(const float4*)(W + n * IN_FEAT + kc + q * 4);
            *(float4*)(&Wlds[n * 68 + q * 4]) = wv;
        }
        __syncthreads();

        // ---- software-pipelined fragment loads ----
        v2f af = *(const v2f*)(xrow + kc + 2 * half);
        v2f bf[8];
#pragma unroll
        for (int nt = 0; nt < 8; ++nt)
            bf[nt] = *(const v2f*)(&Wlds[(nt * 16 + lane16) * 68 + 2 * half]);

#pragma unroll
        for (int kk = 0; kk < 64; kk += 4) {
            v2f an = af;
            v2f bn[8];
#pragma unroll
            for (int nt = 0; nt < 8; ++nt) bn[nt] = bf[nt];
            if (kk + 4 < 64) {
                // prefetch next k-step while WMMAs below run on current regs
                an = *(const v2f*)(xrow + kc + kk + 4 + 2 * half);
#pragma unroll
                for (int nt = 0; nt < 8; ++nt)
                    bn[nt] = *(const v2f*)(&Wlds[(nt * 16 + lane16) * 68 + kk + 4 + 2 * half]);
            }
#pragma unroll
            for (int nt = 0; nt < 8; ++nt) {
                acc[nt] = __builtin_amdgcn_wmma_f32_16x16x4_f32(
                    false, af, false, bf[nt], (short)0, acc[nt], false, false);
            }
            af = an;
#pragma unroll
            for (int nt = 0; nt < 8; ++nt) bf[nt] = bn[nt];
        }
    }

    // C/D layout: VGPR v, lanes 0-15 -> M=v, lanes 16-31 -> M=8+v; N = lane%16
#pragma unroll
    for (int nt = 0; nt < 8; ++nt) {
#pragma unroll
        for (int v = 0; v < 8; ++v) {
            int row = m0 + v + 8 * half;
            if (row < KNODES)
                Wh[(size_t)row * OUT_F + nt * 16 + lane16] = acc[nt][v];
        }
    }
}

// -----------------------------------------------------------------------------
// Kernel 2: per-node attention scores
//   s_src[k,h] = dot(Wh[k,h,:], a[h, 0:32]);  s_dst[k,h] = dot(Wh[k,h,:], a[h,32:64])
// -----------------------------------------------------------------------------
__global__ void gat_node_scores(const float* __restrict__ Wh,
                                const float* __restrict__ a,
                                float* __restrict__ s_src,
                                float* __restrict__ s_dst)
{
    int kh = blockIdx.x * blockDim.x + threadIdx.x;
    if (kh >= KNODES * NHEADS) return;
    int k = kh >> 2, h = kh & 3;
    const float4* row  = (const float4*)(Wh + (size_t)k * OUT_F + h * HDIM);
    const float4* asrc = (const float4*)(a + h * 2 * HDIM);
    const float4* adst = (const float4*)(a + h * 2 * HDIM + HDIM);
    float ss = 0.f, sd = 0.f;
#pragma unroll
    for (int i = 0; i < HDIM / 4; ++i) {
        float4 w = row[i], u = asrc[i], v = adst[i];
        ss += w.x * u.x + w.y * u.y + w.z * u.z + w.w * u.w;
        sd += w.x * v.x + w.y * v.y + w.z * v.z + w.w * v.w;
    }
    s_src[kh] = ss;
    s_dst[kh] = sd;
}

// -----------------------------------------------------------------------------
// Kernel 3: zero e_max / e_sum tables
// -----------------------------------------------------------------------------
__global__ void gat_zero(float* __restrict__ p, int n)
{
    int i = blockIdx.x * blockDim.x + threadIdx.x;
    if (i < n) p[i] = 0.0f;
}

__device__ __forceinline__ float4 edge_logits(const float* __restrict__ s_src,
                                              const float* __restrict__ s_dst,
                                              long long s, long long d, float w)
{
    float4 vs = *(const float4*)(s_src + 4 * (size_t)s);
    float4 vd = *(const float4*)(s_dst + 4 * (size_t)d);
    float4 v;
    v.x = vs.x + vd.x; v.y = vs.y + vd.y; v.z = vs.z + vd.z; v.w = vs.w + vd.w;
    v.x = (v.x > 0.f ? v.x : NEG_SLOPE * v.x) * w;
    v.y = (v.y > 0.f ? v.y : NEG_SLOPE * v.y) * w;
    v.z = (v.z > 0.f ? v.z : NEG_SLOPE * v.z) * w;
    v.w = (v.w > 0.f ? v.w : NEG_SLOPE * v.w) * w;
    return v;
}

// -----------------------------------------------------------------------------
// Kernel 4: scatter-max of edge logits into e_max[dst,h] (init 0, matching
// reference zeros.at[dst].max). e_max >= 0, so int-bit atomicMax is exact for
// positive candidates; non-positive candidates cannot beat the 0 init.
// -----------------------------------------------------------------------------
__global__ void gat_edge_max(const long long* __restrict__ ei,
                             const float* __restrict__ ew,
                             const float* __restrict__ s_src,
                             const float* __restrict__ s_dst,
                             int* __restrict__ e_max_bits)
{
    int e = blockIdx.x * blockDim.x + threadIdx.x;
    if (e >= EEDGES) return;
    long long s = ei[e], d = ei[EEDGES + e];
    float4 v = edge_logits(s_src, s_dst, s, d, ew[e]);
    int* base = e_max_bits + 4 * (size_t)d;
    if (v.x > 0.f) atomicMax(base + 0, __float_as_int(v.x));
    if (v.y > 0.f) atomicMax(base + 1, __float_as_int(v.y));
    if (v.z > 0.f) atomicMax(base + 2, __float_as_int(v.z));
    if (v.w > 0.f) atomicMax(base + 3, __float_as_int(v.w));
}

// -----------------------------------------------------------------------------
// Kernel 5: e_sum[dst,h] += exp(e - e_max[dst,h])   (float atomic add in L2)
// -----------------------------------------------------------------------------
__global__ void gat_edge_sum(const long long* __restrict__ ei,
                             const float* __restrict__ ew,
                             const float* __restrict__ s_src,
                             const float* __restrict__ s_dst,
                             const float* __restrict__ e_max,
                             float* __restrict__ e_sum)
{
    int e = blockIdx.x * blockDim.x + threadIdx.x;
    if (e >= EEDGES) return;
    long long s = ei[e], d = ei[EEDGES + e];
    float4 v  = edge_logits(s_src, s_dst, s, d, ew[e]);
    float4 em = *(const float4*)(e_max + 4 * (size_t)d);
    float* base = e_sum + 4 * (size_t)d;
    atomicAdd(base + 0, expf(v.x - em.x));
    atomicAdd(base + 1, expf(v.y - em.y));
    atomicAdd(base + 2, expf(v.z - em.z));
    atomicAdd(base + 3, expf(v.w - em.w));
}

// -----------------------------------------------------------------------------
// Kernel 6: out = elu( Wh * e_sum/(e_sum + 1e-8) ), in place on d_out.
// (Σ alpha over a dst group collapses to e_sum/(e_sum+1e-8) since Wh_dst is
//  constant within the group.)
// -----------------------------------------------------------------------------
__global__ void gat_finalize(float* __restrict__ out, const float* __restrict__ e_sum)
{
    int q = blockIdx.x * blockDim.x + threadIdx.x;          // float4 index
    if (q >= KNODES * (OUT_F / 4)) return;
    int k = q / (OUT_F / 4);
    int h = (q % (OUT_F / 4)) / (HDIM / 4);
    float es = e_sum[(size_t)k * NHEADS + h];
    float S  = es / (es + 1e-8f);
    float4 w = ((float4*)out)[q];
    w.x *= S; w.y *= S; w.z *= S; w.w *= S;
    w.x = w.x > 0.f ? w.x : expf(w.x) - 1.f;
    w.y = w.y > 0.f ? w.y : expf(w.y) - 1.f;
    w.z = w.z > 0.f ? w.z : expf(w.z) - 1.f;
    w.w = w.w > 0.f ? w.w : expf(w.w) - 1.f;
    ((float4*)out)[q] = w;
}

extern "C" void kernel_launch(void* const* d_in, const int* in_sizes, int n_in,
                              void* d_out, int out_size, void* d_ws, size_t ws_size,
                              hipStream_t stream)
{
    const float*     x  = (const float*)d_in[0];
    const long long* ei = (const long long*)d_in[1];   // int64 edge_index [2,E]
    const float*     ew = (const float*)d_in[2];
    const float*     W  = (const float*)d_in[3];       // [128,256]
    const float*     a  = (const float*)d_in[4];       // [4,64]
    float* out = (float*)d_out;                        // Wh lives here, finalized in place

    const size_t KH = (size_t)KNODES * NHEADS;
    float* s_src = (float*)d_ws;
    float* s_dst = s_src + KH;
    float* e_max = s_dst + KH;
    float* e_sum = e_max + KH;

    // 1) GEMM (WMMA): Wh -> d_out
    gat_gemm_wmma<<<(KNODES + 127) / 128, 256, 0, stream>>>(x, W, out);
    // 2) per-node scores
    gat_node_scores<<<(int)((KH + 255) / 256), 256, 0, stream>>>(out, a, s_src, s_dst);
    // 3) zero e_max & e_sum (contiguous 2*KH floats)
    gat_zero<<<(int)((2 * KH + 255) / 256), 256, 0, stream>>>(e_max, (int)(2 * KH));
    // 4) scatter max
    gat_edge_max<<<(EEDGES + 255) / 256, 256, 0, stream>>>(ei, ew, s_src, s_dst, (int*)e_max);
    // 5) scatter exp-sum
    gat_edge_sum<<<(EEDGES + 255) / 256, 256, 0, stream>>>(ei, ew, s_src, s_dst, e_max, e_sum);
    // 6) scale + ELU in place
    gat_finalize<<<(KNODES * (OUT_F / 4) + 255) / 256, 256, 0, stream>>>(out, e_sum);
}